// GCNEncoder_87608742904438
// MI455X (gfx1250) — compile-verified
//
#include <hip/hip_runtime.h>

typedef float v2f __attribute__((ext_vector_type(2)));
typedef float v8f __attribute__((ext_vector_type(8)));

#define N_NODES 50000
#define N_EDGES 1600000
#define CH      128
#define NGRAPH  64

// ---------------------------------------------------------------------------
// K1: init deg with self-loop weight 1.0, zero pooled sums / counts
// ---------------------------------------------------------------------------
__global__ void k_init(float* __restrict__ deg, float* __restrict__ psum,
                       float* __restrict__ pcnt) {
    int t = blockIdx.x * blockDim.x + threadIdx.x;
    if (t < N_NODES) deg[t] = 1.0f;
    if (t < NGRAPH * CH) psum[t] = 0.0f;
    if (t < NGRAPH) pcnt[t] = 0.0f;
}

// ---------------------------------------------------------------------------
// K2: deg[col[e]] += ew[e]
// ---------------------------------------------------------------------------
__global__ void k_deg(const long long* __restrict__ col,
                      const float* __restrict__ ew,
                      float* __restrict__ deg) {
    int e = blockIdx.x * blockDim.x + threadIdx.x;
    if (e < N_EDGES) atomicAdd(&deg[(int)col[e]], ew[e]);
}

// ---------------------------------------------------------------------------
// K3: dis = rsqrt(deg) (deg>0 guard) + accumulate per-graph node counts
// ---------------------------------------------------------------------------
__global__ void k_dis_count(const float* __restrict__ deg,
                            float* __restrict__ dis,
                            const long long* __restrict__ batch,
                            float* __restrict__ pcnt) {
    int n = blockIdx.x * blockDim.x + threadIdx.x;
    if (n < N_NODES) {
        float d = deg[n];
        dis[n] = (d > 0.0f) ? rsqrtf(d) : 0.0f;
        atomicAdd(&pcnt[(int)batch[n]], 1.0f);
    }
}

// ---------------------------------------------------------------------------
// K4/K8: Y[N,128] = X[N,128] @ W[128,128] via V_WMMA_F32_16X16X4_F32.
// W staged in LDS (64 KB). Each wave: one 16-row stripe, all 8 col tiles.
// A layout (16x4 f32):  lanes 0-15 row M, v0=K0 v1=K1; lanes 16-31 v0=K2 v1=K3.
// B layout (4x16 f32):  v0: K0(lanes0-15)/K2(lanes16-31); v1: K1/K3, N=lane&15.
// C/D layout: VGPR j -> M=j (lanes0-15), M=j+8 (lanes16-31), N=lane&15.
// ---------------------------------------------------------------------------
__global__ __launch_bounds__(256) void k_gemm(const float* __restrict__ X,
                                              const float* __restrict__ Wm,
                                              float* __restrict__ Y,
                                              int nRows) {
    __shared__ float sW[CH * CH];
    int tid = threadIdx.x;
    for (int i = tid; i < CH * CH; i += 256) sW[i] = Wm[i];
    __syncthreads();

    int wid  = tid >> 5;
    int lane = tid & 31;
    int tile = blockIdx.x * 8 + wid;
    int nTiles = nRows >> 4;          // nRows multiple of 16 (50000 = 3125*16)
    if (tile >= nTiles) return;

    int mrow = lane & 15;             // A row / B-D column within tile
    int koff = (lane >> 4) << 1;      // 0 or 2
    int hi8  = (lane >> 4) << 3;      // 0 or 8 (C/D row offset)

    const float* xrow = X + (size_t)(tile * 16 + mrow) * CH;

    v8f c[8];
#pragma unroll
    for (int t = 0; t < 8; ++t) c[t] = (v8f)(0.0f);

#pragma unroll 4
    for (int k0 = 0; k0 < CH; k0 += 4) {
        v2f a;
        a.x = xrow[k0 + koff];
        a.y = xrow[k0 + koff + 1];
        const float* w0 = &sW[(k0 + koff) * CH + mrow];
#pragma unroll
        for (int t = 0; t < 8; ++t) {
            v2f b;
            b.x = w0[t * 16];         // row k0+koff,   col t*16+mrow
            b.y = w0[CH + t * 16];    // row k0+koff+1, col t*16+mrow
            c[t] = __builtin_amdgcn_wmma_f32_16x16x4_f32(
                false, a, false, b, (short)0, c[t], false, false);
        }
    }

#pragma unroll
    for (int t = 0; t < 8; ++t) {
        float* yb = Y + (size_t)(tile * 16 + hi8) * CH + t * 16 + mrow;
#pragma unroll
        for (int j = 0; j < 8; ++j) yb[(size_t)j * CH] = c[t][j];
    }
}

// ---------------------------------------------------------------------------
// K5/K9: self-loop contribution, h[n,f] = xw[n,f] * dis[n]^2 (also clears dst)
// ---------------------------------------------------------------------------
__global__ void k_selfloop(const float* __restrict__ xw,
                           const float* __restrict__ dis,
                           float* __restrict__ h) {
    int t = blockIdx.x * blockDim.x + threadIdx.x;
    if (t < N_NODES * CH) {
        int n = t >> 7;
        float d = dis[n];
        h[t] = xw[t] * d * d;
    }
}

// ---------------------------------------------------------------------------
// K6/K10: edge scatter. One wave per edge; each lane: float4 gather + 4
// global float atomics.  norm = dis[r] * w * dis[c].
// ---------------------------------------------------------------------------
__global__ void k_scatter(const long long* __restrict__ row,
                          const long long* __restrict__ col,
                          const float* __restrict__ ew,
                          const float* __restrict__ dis,
                          const float* __restrict__ src,
                          float* __restrict__ dst) {
    int t = blockIdx.x * blockDim.x + threadIdx.x;   // E*32 = 51.2M < 2^31
    int e = t >> 5;
    if (e >= N_EDGES) return;
    int lane = t & 31;
    int r = (int)row[e];
    int c = (int)col[e];
    float norm = dis[r] * ew[e] * dis[c];
    const float4 v = *(const float4*)(src + (size_t)r * CH + lane * 4);
    float* d = dst + (size_t)c * CH + lane * 4;
    atomicAdd(d + 0, v.x * norm);
    atomicAdd(d + 1, v.y * norm);
    atomicAdd(d + 2, v.z * norm);
    atomicAdd(d + 3, v.w * norm);
}

// ---------------------------------------------------------------------------
// K7: h = relu(h + b)
// ---------------------------------------------------------------------------
__global__ void k_bias_relu(float* __restrict__ h, const float* __restrict__ b) {
    int t = blockIdx.x * blockDim.x + threadIdx.x;
    if (t < N_NODES * CH) {
        int f = t & 127;
        h[t] = fmaxf(h[t] + b[f], 0.0f);
    }
}

// ---------------------------------------------------------------------------
// K11: layer-2 finalize: h = relu(h + b), and accumulate pooled sums
// ---------------------------------------------------------------------------
__global__ void k_bias_relu_pool(float* __restrict__ h,
                                 const float* __restrict__ b,
                                 const long long* __restrict__ batch,
                                 float* __restrict__ psum) {
    int t = blockIdx.x * blockDim.x + threadIdx.x;
    if (t < N_NODES * CH) {
        int f = t & 127;
        int n = t >> 7;
        float v = fmaxf(h[t] + b[f], 0.0f);
        h[t] = v;
        atomicAdd(&psum[(int)batch[n] * CH + f], v);
    }
}

// ---------------------------------------------------------------------------
// K12: pooled = psum / max(count, 1)
// ---------------------------------------------------------------------------
__global__ void k_pool_div(const float* __restrict__ psum,
                           const float* __restrict__ pcnt,
                           float* __restrict__ out) {
    int t = blockIdx.x * blockDim.x + threadIdx.x;
    if (t < NGRAPH * CH) out[t] = psum[t] / fmaxf(pcnt[t >> 7], 1.0f);
}

// ---------------------------------------------------------------------------
extern "C" void kernel_launch(void* const* d_in, const int* in_sizes, int n_in,
                              void* d_out, int out_size, void* d_ws, size_t ws_size,
                              hipStream_t stream) {
    const float*     x     = (const float*)d_in[0];
    const long long* eidx  = (const long long*)d_in[1];
    const float*     ew    = (const float*)d_in[2];
    const long long* batch = (const long long*)d_in[3];
    const float*     W1    = (const float*)d_in[4];
    const float*     b1    = (const float*)d_in[5];
    const float*     W2    = (const float*)d_in[6];
    const float*     b2    = (const float*)d_in[7];

    const long long* row = eidx;
    const long long* col = eidx + N_EDGES;

    // workspace layout (floats)
    float* ws   = (float*)d_ws;
    float* deg  = ws;                                  // N
    float* dis  = ws + N_NODES;                        // N
    float* xw   = ws + 2 * (size_t)N_NODES;            // N*128
    float* h1   = xw + (size_t)N_NODES * CH;           // N*128
    float* psum = h1 + (size_t)N_NODES * CH;           // 64*128
    float* pcnt = psum + NGRAPH * CH;                  // 64

    float* hout = (float*)d_out;                       // N*128
    float* pout = hout + (size_t)N_NODES * CH;         // 64*128

    const dim3 B(256);
    const int gN   = (N_NODES + 255) / 256;
    const int gE   = (N_EDGES + 255) / 256;
    const int gNC  = (N_NODES * CH + 255) / 256;
    const int gE32 = (N_EDGES * 32 + 255) / 256;
    const int gGemm = ((N_NODES / 16) + 7) / 8;        // 8 waves/block
    const int gP   = (NGRAPH * CH + 255) / 256;

    // shared normalization (both layers)
    k_init<<<gN, B, 0, stream>>>(deg, psum, pcnt);
    k_deg<<<gE, B, 0, stream>>>(col, ew, deg);
    k_dis_count<<<gN, B, 0, stream>>>(deg, dis, batch, pcnt);

    // layer 1
    k_gemm<<<gGemm, B, 0, stream>>>(x, W1, xw, N_NODES);
    k_selfloop<<<gNC, B, 0, stream>>>(xw, dis, h1);
    k_scatter<<<gE32, B, 0, stream>>>(row, col, ew, dis, xw, h1);
    k_bias_relu<<<gNC, B, 0, stream>>>(h1, b1);

    // layer 2 (xw buffer reused)
    k_gemm<<<gGemm, B, 0, stream>>>(h1, W2, xw, N_NODES);
    k_selfloop<<<gNC, B, 0, stream>>>(xw, dis, hout);
    k_scatter<<<gE32, B, 0, stream>>>(row, col, ew, dis, xw, hout);
    k_bias_relu_pool<<<gNC, B, 0, stream>>>(hout, b2, batch, psum);

    // mean pool
    k_pool_div<<<gP, B, 0, stream>>>(psum, pcnt, pout);
}